// SparseConv2DLayer_15479062134894
// MI455X (gfx1250) — compile-verified
//
#include <hip/hip_runtime.h>
#include <cstdint>
#include <cstddef>

#define IMG_H 4096
#define IMG_W 4096

// Native 2x64-bit vector so __builtin_nontemporal_load emits one B128 NT load.
typedef long long v2ll __attribute__((ext_vector_type(2)));

// ---------------------------------------------------------------------------
// Pass 1: zero the 4096x4096 f32 output with wide B128 stores. Default (RT)
// temporal hint on purpose: these lines should be resident + dirty in the
// 192MB L2 when the scatter atomics arrive, so atomics resolve at L2.
// ---------------------------------------------------------------------------
__global__ __launch_bounds__(256) void zero_out_kernel(float4* __restrict__ out4, int n4) {
    int i = blockIdx.x * blockDim.x + threadIdx.x;
    if (i < n4) {
        out4[i] = make_float4(0.f, 0.f, 0.f, 0.f);
    }
}

// ---------------------------------------------------------------------------
// Pass 2: one thread per sparse point. 9 non-returning global_atomic_add_f32
// per point (center tap fused with bias). Interior fast path is branch-free.
// Streaming inputs (indices/values/mask) are read once -> non-temporal loads
// so they don't evict the output surface from L2.
// ---------------------------------------------------------------------------
__global__ __launch_bounds__(256) void sparse_conv_scatter_kernel(
    const long long* __restrict__ indices,   // N x 2 int64 (y, x)
    const float*     __restrict__ values,    // N f32
    const float*     __restrict__ mask_vals, // N f32
    const float*     __restrict__ w,         // 3x3x1 f32
    const float*     __restrict__ b,         // 1 f32
    float*           __restrict__ out,       // IMG_H x IMG_W f32
    int n)
{
    const int i = blockIdx.x * blockDim.x + threadIdx.x;

    // Uniform weight/bias loads (uniform address -> scalar cache path).
    const float w00 = w[0], w01 = w[1], w02 = w[2];
    const float w10 = w[3], w11 = w[4], w12 = w[5];
    const float w20 = w[6], w21 = w[7], w22 = w[8];
    const float bias = b[0];

    if (i >= n) return;

    // Prefetch the index stream ahead of the atomic issue window
    // (lowers to gfx1250 global_prefetch_b8; guarded so address is valid).
    const int PF_DIST = 2048;
    if (i + PF_DIST < n) {
        __builtin_prefetch(&indices[2 * (size_t)(i + PF_DIST)], 0, 0);
    }

    // Single non-temporal global_load_b128 for (y, x).
    const v2ll yx = __builtin_nontemporal_load((const v2ll*)indices + i);
    const int y = (int)yx.x;
    const int x = (int)yx.y;

    const float v  = __builtin_nontemporal_load(values + i);
    const float mv = __builtin_nontemporal_load(mask_vals + i);
    // Center tap is always in-bounds (indices in [0,4096)), fuse bias into it.
    const float center = v * w11 + mv * bias;

    float* const row = out + (size_t)y * IMG_W + x;

    if (y >= 1 && y < IMG_H - 1 && x >= 1 && x < IMG_W - 1) {
        // Interior fast path: 9 unconditional non-returning f32 atomics (L2-resident).
        atomicAdd(row - IMG_W - 1, v * w00);
        atomicAdd(row - IMG_W    , v * w01);
        atomicAdd(row - IMG_W + 1, v * w02);
        atomicAdd(row         - 1, v * w10);
        atomicAdd(row            , center );
        atomicAdd(row         + 1, v * w12);
        atomicAdd(row + IMG_W - 1, v * w20);
        atomicAdd(row + IMG_W    , v * w21);
        atomicAdd(row + IMG_W + 1, v * w22);
    } else {
        // Border slow path (<0.1% of points): per-tap bounds checks.
        const float wk[9] = { w00, w01, w02, w10, w11, w12, w20, w21, w22 };
        #pragma unroll
        for (int ky = 0; ky < 3; ++ky) {
            #pragma unroll
            for (int kx = 0; kx < 3; ++kx) {
                const int ny = y + ky - 1;
                const int nx = x + kx - 1;
                if (ny >= 0 && ny < IMG_H && nx >= 0 && nx < IMG_W) {
                    float contrib = (ky == 1 && kx == 1) ? center : v * wk[ky * 3 + kx];
                    atomicAdd(out + (size_t)ny * IMG_W + nx, contrib);
                }
            }
        }
    }
}

// ---------------------------------------------------------------------------
// Launcher
// ---------------------------------------------------------------------------
extern "C" void kernel_launch(void* const* d_in, const int* in_sizes, int n_in,
                              void* d_out, int out_size, void* d_ws, size_t ws_size,
                              hipStream_t stream)
{
    const long long* indices   = (const long long*)d_in[0]; // N*2 int64
    const float*     values    = (const float*)d_in[1];     // N
    const float*     mask_vals = (const float*)d_in[2];     // N
    const float*     w         = (const float*)d_in[3];     // 9
    const float*     b         = (const float*)d_in[4];     // 1
    float*           out       = (float*)d_out;             // IMG_H*IMG_W

    const int n = in_sizes[1]; // N (values is flat N)

    // Pass 1: zero output (out_size = 16,777,216 floats = 4,194,304 float4s).
    const int n4 = out_size / 4;
    {
        const int threads = 256;
        const int blocks  = (n4 + threads - 1) / threads;
        zero_out_kernel<<<blocks, threads, 0, stream>>>((float4*)out, n4);
    }

    // Pass 2: scatter-add the sparse convolution.
    {
        const int threads = 256;
        const int blocks  = (n + threads - 1) / threads;
        sparse_conv_scatter_kernel<<<blocks, threads, 0, stream>>>(
            indices, values, mask_vals, w, b, out, n);
    }
}